// CoordinateDensification_31739808317552
// MI455X (gfx1250) — compile-verified
//
#include <hip/hip_runtime.h>
#include <stdint.h>

// CoordinateDensification on gfx1250 (MI455X), wave32.
//
// Order-preserving dense key  k = ((z+s)*R + (y+s))*R + (x+s),
// R = 256 + 2*stride (coords in [0,256), offsets in {-s,0,s}).
// Dedup via L2-resident bitmap; sorted compaction via hierarchical
// popcount prefix sum; padding written only for the tail (count known
// on device after the scan).
//
// Workspace layout (uint32 words):
//   [0     , W_MAX)        bitmap (16 MiB; supports stride<=128, R<=512)
//   [W_MAX , W_MAX+NBLK+1) block sums -> exclusive offsets; [NBLK] = total
#define W_MAX   (4194304u)      // 512^3 / 32 words
#define NBLK    (W_MAX / 256u)  // 16384 scan blocks of 256 words
#define FILLV   (2147483647)    // INT32_MAX

// ---------------------------------------------------------------- clear bitmap
__global__ void k_clear(uint32_t* __restrict__ bm) {
  const uint4 z = make_uint4(0u, 0u, 0u, 0u);
  const uint32_t nv = W_MAX >> 2;
  for (uint32_t i = blockIdx.x * blockDim.x + threadIdx.x; i < nv;
       i += gridDim.x * blockDim.x) {
    ((uint4*)bm)[i] = z;
  }
}

// ------------------------------------------------------- expand + set key bits
// Coordinate rows staged through LDS with the gfx1250 async global->LDS path.
__global__ void __launch_bounds__(256)
k_set_bits(const int* __restrict__ coords, const int* __restrict__ stride_p,
           uint32_t* __restrict__ bm, int n) {
  __shared__ int4 tile[256];

  const int s = *stride_p;                   // uniform scalar
  const uint32_t us    = (uint32_t)s;
  const uint32_t two_s = 2u * us;
  const uint32_t R     = 256u + two_s;

  const int i   = blockIdx.x * 256 + (int)threadIdx.x;
  const int src = (i < n) ? i : (n - 1);     // valid address in tail block

  {
    // VDST = LDS byte offset (low 32 bits of generic shared pointer per the
    // LDS aperture layout); VADDR = 64-bit global address.
    uint32_t lds_off = (uint32_t)(uintptr_t)(&tile[threadIdx.x]);
    uint64_t gaddr   = (uint64_t)(uintptr_t)(coords + (size_t)src * 4);
    asm volatile("global_load_async_to_lds_b128 %0, %1, off"
                 :: "v"(lds_off), "v"(gaddr) : "memory");
#if __has_builtin(__builtin_amdgcn_s_wait_asynccnt)
    __builtin_amdgcn_s_wait_asynccnt(0);
#else
    asm volatile("s_wait_asynccnt 0" ::: "memory");
#endif
  }
  __syncthreads();

  if (i >= n) return;

  const int4 c = tile[threadIdx.x];          // [c0,c1,c2,0], c0 most significant
  const uint32_t b0 = (uint32_t)(c.x + s);   // biased, in [0, R-1]
  const uint32_t b1 = (uint32_t)(c.y + s);
  const uint32_t b2 = (uint32_t)(c.z + s);

  #pragma unroll
  for (int dz = 0; dz < 3; ++dz) {
    const uint32_t kz = (b0 + (uint32_t)dz * us - us) * R;
    #pragma unroll
    for (int dy = 0; dy < 3; ++dy) {
      const uint32_t kzy = (kz + b1 + (uint32_t)dy * us - us) * R;
      // three x-offsets: keys k0, k0+s, k0+2s — usually in one word
      const uint32_t k0  = kzy + b2 - us;
      const uint32_t low = k0 & 31u;
      if (low + two_s < 32u) {
        const uint32_t mask = (1u | (1u << us) | (1u << two_s)) << low;
        atomicOr(&bm[k0 >> 5], mask);
      } else {
        atomicOr(&bm[k0 >> 5], 1u << low);
        const uint32_t k1 = k0 + us;
        atomicOr(&bm[k1 >> 5], 1u << (k1 & 31u));
        const uint32_t k2 = k0 + two_s;
        atomicOr(&bm[k2 >> 5], 1u << (k2 & 31u));
      }
    }
  }
}

// -------------------------------------------------- per-block popcount reduce
__global__ void __launch_bounds__(256)
k_scan1(const uint32_t* __restrict__ bm, uint32_t* __restrict__ blk) {
  __shared__ uint32_t sh[256];
  const uint32_t i = blockIdx.x * 256u + threadIdx.x;
  sh[threadIdx.x] = (uint32_t)__popc(bm[i]);
  __syncthreads();
  #pragma unroll
  for (int off = 128; off > 0; off >>= 1) {
    if ((int)threadIdx.x < off) sh[threadIdx.x] += sh[threadIdx.x + off];
    __syncthreads();
  }
  if (threadIdx.x == 0) blk[blockIdx.x] = sh[0];
}

// ------------------- single-block exclusive scan (in place); blk[NBLK] = total
__global__ void __launch_bounds__(256)
k_scan2(uint32_t* __restrict__ blk) {
  __shared__ uint32_t sh[256];
  uint32_t carry = 0;
  for (uint32_t base = 0; base < NBLK; base += 256u) {
    const uint32_t v = blk[base + threadIdx.x];
    sh[threadIdx.x] = v;
    __syncthreads();
    #pragma unroll
    for (int off = 1; off < 256; off <<= 1) {
      const uint32_t t = ((int)threadIdx.x >= off) ? sh[threadIdx.x - off] : 0u;
      __syncthreads();
      sh[threadIdx.x] += t;
      __syncthreads();
    }
    const uint32_t incl  = sh[threadIdx.x];
    const uint32_t total = sh[255];
    blk[base + threadIdx.x] = carry + incl - v;   // exclusive prefix
    carry += total;
    __syncthreads();
  }
  if (threadIdx.x == 0) blk[NBLK] = carry;        // total unique rows
}

// --------------------------------------------- emit sorted unique rows (int4)
__global__ void __launch_bounds__(256)
k_emit(const uint32_t* __restrict__ bm, const uint32_t* __restrict__ blk,
       const int* __restrict__ stride_p, int* __restrict__ out) {
  __shared__ uint32_t sh[256];
  const int s = *stride_p;
  const uint32_t R = 256u + 2u * (uint32_t)s;

  const uint32_t i = blockIdx.x * 256u + threadIdx.x;
  if (i + 8192u < W_MAX)                    // stream-ahead (global_prefetch_b8)
    __builtin_prefetch(bm + i + 8192u, 0, 0);

  uint32_t w = bm[i];
  const uint32_t c = (uint32_t)__popc(w);
  sh[threadIdx.x] = c;
  __syncthreads();
  #pragma unroll
  for (int off = 1; off < 256; off <<= 1) {
    const uint32_t t = ((int)threadIdx.x >= off) ? sh[threadIdx.x - off] : 0u;
    __syncthreads();
    sh[threadIdx.x] += t;
    __syncthreads();
  }
  uint32_t pos = blk[blockIdx.x] + sh[threadIdx.x] - c;   // output row index

  if (!w) return;

  // Decode base (z,y,x) once per word; within a word keys cross at most one
  // x-row boundary since 32 <= R.
  const uint32_t keyBase = i * 32u;
  uint32_t q  = keyBase / R;
  int32_t  xb = (int32_t)(keyBase - q * R);   // keyBase % R
  uint32_t z  = q / R;
  uint32_t y  = q - z * R;                    // q % R

  while (w) {
    const uint32_t b = (uint32_t)__ffs(w) - 1u;  // ascending bits => sorted keys
    w &= (w - 1u);
    int32_t x = xb + (int32_t)b;
    if (x >= (int32_t)R) {                    // crossed into next x-row
      xb -= (int32_t)R;
      x  -= (int32_t)R;
      if (++y == R) { y = 0u; ++z; }
    }
    ((int4*)out)[pos++] =
        make_int4((int)z - s, (int)y - s, x - s, 0);
  }
}

// --------------------------- write INT32_MAX padding rows for [total, nrows)
__global__ void k_fill_tail(int* __restrict__ out,
                            const uint32_t* __restrict__ total_p, int nrows) {
  const uint32_t total = *total_p;            // uniform scalar
  const int4 f = make_int4(FILLV, FILLV, FILLV, FILLV);
  for (uint32_t row = total + blockIdx.x * blockDim.x + threadIdx.x;
       row < (uint32_t)nrows; row += gridDim.x * blockDim.x) {
    ((int4*)out)[row] = f;
  }
}

// ------------------------------------------------------------------- launcher
extern "C" void kernel_launch(void* const* d_in, const int* in_sizes, int n_in,
                              void* d_out, int out_size, void* d_ws, size_t ws_size,
                              hipStream_t stream) {
  const int* d_coords = (const int*)d_in[0];
  const int* d_stride = (const int*)d_in[1];
  int*       out      = (int*)d_out;

  const int N     = in_sizes[0] / 4;          // rows of [z,y,x,b]
  const int NROWS = out_size / 4;             // N * 27 output rows

  uint32_t* bitmap = (uint32_t*)d_ws;
  uint32_t* blk    = bitmap + W_MAX;          // NBLK+1 words

  // 1. clear bitmap
  k_clear<<<4096, 256, 0, stream>>>(bitmap);
  // 2. neighbor expansion -> bitmap (dedup via atomicOr, merged x-masks)
  k_set_bits<<<(N + 255) / 256, 256, 0, stream>>>(d_coords, d_stride, bitmap, N);
  // 3. popcount block sums
  k_scan1<<<NBLK, 256, 0, stream>>>(bitmap, blk);
  // 4. exclusive scan of block sums (+ total count)
  k_scan2<<<1, 256, 0, stream>>>(blk);
  // 5. sorted compaction into output rows [0, total)
  k_emit<<<NBLK, 256, 0, stream>>>(bitmap, blk, d_stride, out);
  // 6. padding rows [total, NROWS) only
  k_fill_tail<<<(NROWS + 255) / 256, 256, 0, stream>>>(
      out, blk + NBLK, NROWS);
}